// GAT_v2_23055384445760
// MI455X (gfx1250) — compile-verified
//
#include <hip/hip_runtime.h>
#include <hip/hip_bf16.h>

typedef float v2f __attribute__((ext_vector_type(2)));
typedef float v8f __attribute__((ext_vector_type(8)));

#define NEG_SLOPE 0.2f

// ---------------------------------------------------------------------------
// Order-preserving float <-> uint mapping so segment-max can use native
// GLOBAL_ATOMIC_MAX_U32 (monotone: f1 < f2  <=>  key(f1) < key(f2)).
// ---------------------------------------------------------------------------
__device__ __forceinline__ unsigned f2key(float f) {
  unsigned b = __float_as_uint(f);
  return (b & 0x80000000u) ? ~b : (b | 0x80000000u);
}
__device__ __forceinline__ float key2f(unsigned k) {
  return (k & 0x80000000u) ? __uint_as_float(k & 0x7fffffffu)
                           : __uint_as_float(~k);
}

// ---------------------------------------------------------------------------
// C[N,M] = A[N,128] * B[128,M], fp32, one wave per 16x16 tile,
// 32 chained V_WMMA_F32_16X16X4_F32 per tile.
// A-frag (16x4 MxK): lanes 0-15 -> K = k,k+1 ; lanes 16-31 -> K = k+2,k+3.
// B-frag (4x16 KxN): VGPR0 rows K=k (lo lanes) / K=k+2 (hi lanes), VGPR1 +1.
// C/D (16x16): VGPR r -> M = r (lanes 0-15) / M = r+8 (lanes 16-31), N = lane%16.
// Requires N % 16 == 0, M % 16 == 0 (true here: 100000, 128, 160).
// ---------------------------------------------------------------------------
__global__ void gemm_f32_wmma(const float* __restrict__ A,
                              const float* __restrict__ B,
                              float* __restrict__ C, int N, int M) {
  constexpr int K = 128;
  int gid  = blockIdx.x * blockDim.x + threadIdx.x;
  int wave = gid >> 5;
  int lane = threadIdx.x & 31;
  int tiles_n = M >> 4;
  int tm = wave / tiles_n;
  int tn = wave - tm * tiles_n;
  if (tm * 16 >= N) return;               // uniform per-wave exit (EXEC stays all-1s)
  int half = lane >> 4;                   // 0: lanes 0-15, 1: lanes 16-31
  int l    = lane & 15;
  const float* Arow = A + (size_t)(tm * 16 + l) * K + 2 * half;
  const float* Bcol = B + (size_t)(2 * half) * M + tn * 16 + l;
  v8f acc = {};
#pragma unroll
  for (int k = 0; k < K; k += 4) {
    v2f a = *(const v2f*)(Arow + k);      // A[row][k+2h .. k+2h+1], 8B aligned
    v2f b;
    b.x = Bcol[(size_t)k * M];            // B[k+2h  ][col]
    b.y = Bcol[(size_t)(k + 1) * M];      // B[k+2h+1][col]
    acc = __builtin_amdgcn_wmma_f32_16x16x4_f32(
        /*neg_a=*/false, a, /*neg_b=*/false, b,
        /*c_mod=*/(short)0, acc, /*reuse_a=*/false, /*reuse_b=*/false);
  }
  float* Crow = C + (size_t)(tm * 16 + 8 * half) * M + tn * 16 + l;
#pragma unroll
  for (int r = 0; r < 8; ++r) Crow[(size_t)r * M] = acc[r];
}

// ---------------------------------------------------------------------------
// Fills
// ---------------------------------------------------------------------------
__global__ void fill_f32_kernel(float* __restrict__ p, float v, size_t n) {
  size_t i = blockIdx.x * (size_t)blockDim.x + threadIdx.x;
  if (i < n) p[i] = v;
}
__global__ void fill_u32_kernel(unsigned* __restrict__ p, unsigned v, size_t n) {
  size_t i = blockIdx.x * (size_t)blockDim.x + threadIdx.x;
  if (i < n) p[i] = v;
}

// ---------------------------------------------------------------------------
// al_s[n,h] = sum_c h[n,h,c]*a_src[h,c] ; al_d likewise. One thread per (n,h).
// h is [N,H,C] flat, so row base = i*C for i = n*H + h.
// ---------------------------------------------------------------------------
__global__ void attn_logits(const float* __restrict__ h,
                            const float* __restrict__ a_src,
                            const float* __restrict__ a_dst,
                            float* __restrict__ al_s, float* __restrict__ al_d,
                            int NH, int H, int C) {
  int i = blockIdx.x * blockDim.x + threadIdx.x;
  if (i >= NH) return;
  int hd = i % H;
  const float* hp = h + (size_t)i * C;
  const float* asr = a_src + (size_t)hd * C;
  const float* adr = a_dst + (size_t)hd * C;
  float ss = 0.f, sd = 0.f;
  for (int c = 0; c < C; ++c) {
    float v = hp[c];
    ss += v * asr[c];
    sd += v * adr[c];
  }
  al_s[i] = ss;
  al_d[i] = sd;
}

// ---------------------------------------------------------------------------
// e = leaky_relu(al_s[src] + al_d[dst]); segment-max into mkey via u32 atomics.
// One thread per (edge, head).
// ---------------------------------------------------------------------------
__global__ void edge_logits_max(const int* __restrict__ src,
                                const int* __restrict__ dst,
                                const float* __restrict__ al_s,
                                const float* __restrict__ al_d,
                                float* __restrict__ ebuf,
                                unsigned* __restrict__ mkey, int E, int H) {
  int i = blockIdx.x * blockDim.x + threadIdx.x;
  if (i >= E * H) return;
  int e = i / H, hd = i - e * H;
  int s = src[e], d = dst[e];
  float v = al_s[s * H + hd] + al_d[d * H + hd];
  v = v > 0.f ? v : NEG_SLOPE * v;
  ebuf[i] = v;
  atomicMax(&mkey[d * H + hd], f2key(v));
}

// ex = exp(e - m[dst]); segment-sum into ssum. One thread per (edge, head).
__global__ void edge_exp_sum(const int* __restrict__ dst,
                             float* __restrict__ ebuf,
                             const unsigned* __restrict__ mkey,
                             float* __restrict__ ssum, int E, int H) {
  int i = blockIdx.x * blockDim.x + threadIdx.x;
  if (i >= E * H) return;
  int e = i / H, hd = i - e * H;
  int d = dst[e];
  float ex = __expf(ebuf[i] - key2f(mkey[d * H + hd]));
  ebuf[i] = ex;
  atomicAdd(&ssum[d * H + hd], ex);
}

// alpha = ex / (s[dst] + 1e-16). One thread per (edge, head).
__global__ void edge_alpha(const int* __restrict__ dst,
                           float* __restrict__ ebuf,
                           const float* __restrict__ ssum, int E, int H) {
  int i = blockIdx.x * blockDim.x + threadIdx.x;
  if (i >= E * H) return;
  int e = i / H, hd = i - e * H;
  ebuf[i] = ebuf[i] / (ssum[dst[e] * H + hd] + 1e-16f);
}

// out[dst,h,c] += h[src,h,c] * alpha[e,h]. One thread per (edge, head, chan).
__global__ void edge_scatter(const int* __restrict__ src,
                             const int* __restrict__ dst,
                             const float* __restrict__ alpha,
                             const float* __restrict__ h,
                             float* __restrict__ out, int E, int H, int C) {
  size_t i = blockIdx.x * (size_t)blockDim.x + threadIdx.x;
  size_t total = (size_t)E * H * C;
  if (i >= total) return;
  int c  = (int)(i % C);
  size_t eh = i / C;
  int hd = (int)(eh % H);
  int e  = (int)(eh / H);
  int s = src[e], d = dst[e];
  float v = h[((size_t)s * H + hd) * C + c] * alpha[eh];
  atomicAdd(&out[((size_t)d * H + hd) * C + c], v);
}

// x = relu(x + b[i % M]), in place.
__global__ void bias_relu(float* __restrict__ x, const float* __restrict__ b,
                          size_t total, int M) {
  size_t i = blockIdx.x * (size_t)blockDim.x + threadIdx.x;
  if (i >= total) return;
  float v = x[i] + b[(int)(i % M)];
  x[i] = v > 0.f ? v : 0.f;
}

// y[n,:] = log_softmax(mean_h(acc[n,h,:]) + b). One thread per node (C<=64).
__global__ void mean_bias_logsoftmax(const float* __restrict__ acc,
                                     const float* __restrict__ b,
                                     float* __restrict__ y, int N, int H, int C) {
  int n = blockIdx.x * blockDim.x + threadIdx.x;
  if (n >= N) return;
  float v[64];
  float invH = 1.0f / (float)H;
  float mx = -3.4e38f;
  for (int c = 0; c < C; ++c) {
    float s = 0.f;
    for (int hd = 0; hd < H; ++hd) s += acc[((size_t)n * H + hd) * C + c];
    float t = s * invH + b[c];
    v[c] = t;
    mx = fmaxf(mx, t);
  }
  float sum = 0.f;
  for (int c = 0; c < C; ++c) sum += __expf(v[c] - mx);
  float lse = mx + __logf(sum);
  for (int c = 0; c < C; ++c) y[(size_t)n * C + c] = v[c] - lse;
}

// ---------------------------------------------------------------------------
static inline int cdiv(long long a, long long b) { return (int)((a + b - 1) / b); }

extern "C" void kernel_launch(void* const* d_in, const int* in_sizes, int n_in,
                              void* d_out, int out_size, void* d_ws, size_t ws_size,
                              hipStream_t stream) {
  (void)n_in; (void)out_size; (void)ws_size;
  const float* x   = (const float*)d_in[0];
  const int*   ei  = (const int*)d_in[1];
  const float* W1  = (const float*)d_in[2];
  const float* as1 = (const float*)d_in[3];
  const float* ad1 = (const float*)d_in[4];
  const float* b1  = (const float*)d_in[5];
  const float* W2  = (const float*)d_in[6];
  const float* as2 = (const float*)d_in[7];
  const float* ad2 = (const float*)d_in[8];
  const float* b2  = (const float*)d_in[9];

  const int F_IN = 128, H = 4, HID = 32, NCLS = 40;
  const int M1 = H * HID;   // 128
  const int M2 = H * NCLS;  // 160
  const int N = in_sizes[0] / F_IN;
  const int E = in_sizes[1] / 2;
  const int* src = ei;       // edge_index[0]
  const int* dst = ei + E;   // edge_index[1]

  // ---- workspace carve-up (floats) ----
  float* ws = (float*)d_ws;
  size_t o = 0;
  float*    buf_h  = ws + o; o += (size_t)N * M2;  // h1 (N*128) then h2 (N*160)
  float*    buf_o1 = ws + o; o += (size_t)N * M1;  // layer-1 aggregated output
  float*    buf_o2 = ws + o; o += (size_t)N * M2;  // layer-2 aggregated output
  float*    ebuf   = ws + o; o += (size_t)E * H;   // edge logits -> ex -> alpha
  float*    al_s   = ws + o; o += (size_t)N * H;
  float*    al_d   = ws + o; o += (size_t)N * H;
  unsigned* mkey   = (unsigned*)(ws + o); o += (size_t)N * H;
  float*    ssum   = ws + o; o += (size_t)N * H;

  const int B = 256;
  const int NH = N * H;
  const long long EH = (long long)E * H;

  // ================= Layer 1 =================
  {
    int waves = (N / 16) * (M1 / 16);
    gemm_f32_wmma<<<cdiv((long long)waves * 32, B), B, 0, stream>>>(x, W1, buf_h, N, M1);
  }
  attn_logits<<<cdiv(NH, B), B, 0, stream>>>(buf_h, as1, ad1, al_s, al_d, NH, H, HID);

  fill_u32_kernel<<<cdiv(NH, B), B, 0, stream>>>(mkey, 0u, (size_t)NH);
  fill_f32_kernel<<<cdiv(NH, B), B, 0, stream>>>(ssum, 0.f, (size_t)NH);
  fill_f32_kernel<<<cdiv((long long)N * M1, B), B, 0, stream>>>(buf_o1, 0.f, (size_t)N * M1);

  edge_logits_max<<<cdiv(EH, B), B, 0, stream>>>(src, dst, al_s, al_d, ebuf, mkey, E, H);
  edge_exp_sum<<<cdiv(EH, B), B, 0, stream>>>(dst, ebuf, mkey, ssum, E, H);
  edge_alpha<<<cdiv(EH, B), B, 0, stream>>>(dst, ebuf, ssum, E, H);
  edge_scatter<<<cdiv(EH * HID, B), B, 0, stream>>>(src, dst, ebuf, buf_h, buf_o1, E, H, HID);

  bias_relu<<<cdiv((long long)N * M1, B), B, 0, stream>>>(buf_o1, b1, (size_t)N * M1, M1);

  // ================= Layer 2 =================
  {
    int waves = (N / 16) * (M2 / 16);
    gemm_f32_wmma<<<cdiv((long long)waves * 32, B), B, 0, stream>>>(buf_o1, W2, buf_h, N, M2);
  }
  attn_logits<<<cdiv(NH, B), B, 0, stream>>>(buf_h, as2, ad2, al_s, al_d, NH, H, NCLS);

  fill_u32_kernel<<<cdiv(NH, B), B, 0, stream>>>(mkey, 0u, (size_t)NH);
  fill_f32_kernel<<<cdiv(NH, B), B, 0, stream>>>(ssum, 0.f, (size_t)NH);
  fill_f32_kernel<<<cdiv((long long)N * M2, B), B, 0, stream>>>(buf_o2, 0.f, (size_t)N * M2);

  edge_logits_max<<<cdiv(EH, B), B, 0, stream>>>(src, dst, al_s, al_d, ebuf, mkey, E, H);
  edge_exp_sum<<<cdiv(EH, B), B, 0, stream>>>(dst, ebuf, mkey, ssum, E, H);
  edge_alpha<<<cdiv(EH, B), B, 0, stream>>>(dst, ebuf, ssum, E, H);
  edge_scatter<<<cdiv(EH * NCLS, B), B, 0, stream>>>(src, dst, ebuf, buf_h, buf_o2, E, H, NCLS);

  mean_bias_logsoftmax<<<cdiv(N, B), B, 0, stream>>>(buf_o2, b2, (float*)d_out, N, H, NCLS);
}